// WindowAttention_37349035606803
// MI455X (gfx1250) — compile-verified
//
#include <hip/hip_runtime.h>
#include <cstdint>
#include <cmath>

typedef float v2f __attribute__((ext_vector_type(2)));
typedef float v8f __attribute__((ext_vector_type(8)));
typedef unsigned int u32x4 __attribute__((ext_vector_type(4)));
typedef int i32x4 __attribute__((ext_vector_type(4)));
typedef int i32x8 __attribute__((ext_vector_type(8)));

#define NX      343
#define NXNX    (343*343)
#define NTOK    344
#define NPAD    352
#define NTILES  22
#define CDIM    96
#define HEADS   3
#define HD      32
#define NWIN    64
#define QSTR    34              // padded row stride for qs/ks (conflict-free)
#define XSTR    99              // xs row stride = 96 + TDM pad (1 DW / 32 DW)
#define X_OUT_ELEMS (512ll*343*96)

// workspace layout (floats)
#define ATTN_WS   0ll
#define BIAS_OFF  (512ll*344*96)            // 16,908,288
#define WQKVF_OFF (BIAS_OFF + 352960)       // 3*343*343 = 352,947 padded
#define WPROJF_OFF (WQKVF_OFF + 27648)      // 3*3*2*24*64
// total ~17.3M floats (~69.2 MB)

__device__ __forceinline__ v8f wmma4(v2f a, v2f b, v8f c) {
  // D = A(16x4,f32) * B(4x16,f32) + C(16x16,f32) -> v_wmma_f32_16x16x4_f32
  return __builtin_amdgcn_wmma_f32_16x16x4_f32(false, a, false, b, (short)0, c,
                                               false, false);
}

// xs element address with TDM intra-row padding (1 DW after every 32 DW)
__device__ __forceinline__ int xsidx(int n, int c) {
  return n * XSTR + c + (c >> 5);
}

// 1-D Tensor-Data-Mover load: ndwords f32 from gptr into LDS at lds_addr.
// flags_w0 carries data_size + optional pad config (D# group1 word0).
// Toolchain uses the 6-arg builtin: (g0 u32x4, g1 i32x8, g2 i32x4, g3 i32x4,
// g4 i32x8, cpol) — extra groups zero for <=2-D tensors.
__device__ __forceinline__ void tdm_load_1d(uint32_t lds_addr, const void* gptr,
                                            uint32_t ndwords, uint32_t flags_w0)
{
  uint64_t ga = (uint64_t)(uintptr_t)gptr;
  u32x4 g0;
  g0[0] = 1u;                                            // count=1, user D#
  g0[1] = lds_addr;                                      // LDS byte address
  g0[2] = (uint32_t)ga;                                  // global_addr lo
  g0[3] = (uint32_t)((ga >> 32) & 0x01FFFFFFu) | (2u << 30);  // ga hi | type=2
  i32x8 g1;
  g1[0] = (int)flags_w0;                                 // data_size=4B (+pad)
  g1[1] = (int)((ndwords & 0xFFFFu) << 16);              // tensor_dim0[15:0]
  g1[2] = (int)(((ndwords >> 16) & 0xFFFFu) | (1u << 16)); // td0 hi | td1=1
  g1[3] = (int)((ndwords & 0xFFFFu) << 16);              // td1 hi=0 | tile_dim0
  g1[4] = 0;                                             // tile_dim1/2 = 0 (1-D)
  g1[5] = (int)ndwords;                                  // tensor_dim0_stride lo
  g1[6] = 0;
  g1[7] = 0;
  i32x4 z4 = {0, 0, 0, 0};
  i32x8 z8 = {0, 0, 0, 0, 0, 0, 0, 0};
  __builtin_amdgcn_tensor_load_to_lds(g0, g1, z4, z4, z8, 0);
}

// ---------------------------------------------------------------------------
// Kernel 0: one-time prep — gather bias table per head, and re-lay Wqkv/Wproj
// into WMMA B-fragment order (64-float chunk per (tile,kstep): coalesced b64)
// ---------------------------------------------------------------------------
__global__ void wattn_prep_kernel(const int* __restrict__ rel,
                                  const float* __restrict__ btab,
                                  const float* __restrict__ Wqkv,
                                  const float* __restrict__ Wproj,
                                  float* __restrict__ biasHM,
                                  float* __restrict__ wqkvf,
                                  float* __restrict__ wprojf)
{
  int t = blockIdx.x * 256 + threadIdx.x;
  if (t < NXNX) {
    int v = rel[t];
    #pragma unroll
    for (int h = 0; h < HEADS; ++h)
      biasHM[h * NXNX + t] = btab[v * HEADS + h];
  }
  if (t < 3 * HEADS * 2 * 24 * 64) {
    int j = t & 1, lane = (t >> 1) & 31;
    int chunk = t >> 6;
    int kstep = chunk % 24;
    int c2 = chunk / 24;
    int nt = c2 & 1;
    int c3 = c2 >> 1;
    int s = c3 % 3, h = c3 / 3;
    wqkvf[t] = Wqkv[(size_t)(s * CDIM + h * HD + nt * 16 + (lane & 15)) * CDIM
                    + kstep * 4 + 2 * (lane >> 4) + j];
  }
  if (t < 6 * 24 * 64) {
    int j = t & 1, lane = (t >> 1) & 31;
    int chunk = t >> 6;
    int kstep = chunk % 24;
    int nt = chunk / 24;
    wprojf[t] = Wproj[(size_t)(nt * 16 + (lane & 15)) * CDIM
                      + kstep * 4 + 2 * (lane >> 4) + j];
  }
}

// ---------------------------------------------------------------------------
// Kernel 1: per-window fused QKV + attention (TDM-staged x, flash softmax)
// one workgroup (256 thr / 8 waves) per batch window b; writes (B,344,96) ws
// ---------------------------------------------------------------------------
__global__ __launch_bounds__(256, 1)
void wattn_fused_kernel(const float* __restrict__ x, const float* __restrict__ gt,
                        const float* __restrict__ mask,
                        const float* __restrict__ bqkv,
                        const float* __restrict__ biasHM,
                        const float* __restrict__ wqkvf,
                        float* __restrict__ attn_out)
{
  __shared__ float xs[NPAD * XSTR];      // xc, TDM-padded rows (139.4KB)
  __shared__ float qs[NPAD * QSTR];      // 47.9KB (conflict-free stride 34)
  __shared__ float ks[NPAD * QSTR];      // 47.9KB
  __shared__ float vs[NPAD * HD];        // 45.1KB (B-side reads are contiguous)
  __shared__ float pscr[8][16 * 17];     // per-wave P transpose scratch (8.7KB)

  const int tid  = threadIdx.x;
  const int lane = tid & 31;
  const int wv   = tid >> 5;
  const int half = lane >> 4;
  const int l16  = lane & 15;
  const int b    = blockIdx.x;
  const float scale = 0.17677669529663687f;   // 1/sqrt(32)

  // --- stage xc: rows 1..343 via TDM (pad 1 DW per 32 DW -> stride 99) ---
  if (wv == 0) {
    // data_size=4B | pad_enable | pad_interval=32DW(code4) | pad_amount=1DW
    tdm_load_1d((uint32_t)(uintptr_t)&xs[XSTR],
                x + (size_t)b * NX * CDIM,
                (uint32_t)(NX * CDIM), 0x01120000u);
  }
  if (tid < CDIM)                              // row 0 = global token
    xs[xsidx(0, tid)] = gt[(size_t)b * CDIM + tid];
  for (int t = tid; t < 8 * XSTR; t += 256)    // zero pad rows 344..351
    xs[NTOK * XSTR + t] = 0.f;
  if (wv == 0)
    __builtin_amdgcn_s_wait_tensorcnt((short)0);
  __syncthreads();

  const float* maskw = mask + (size_t)(b & (NWIN - 1)) * NXNX;

  for (int h = 0; h < HEADS; ++h) {
    const float* biasH = biasHM + (size_t)h * NXNX;

    // ---- QKV GEMM: {q*scale,k,v}(352x32) = xs(352x96) @ Wqkv_slice^T ----
    for (int job = wv; job < 3 * NTILES * 2; job += 8) {
      int s   = job / (NTILES * 2);
      int rem = job - s * (NTILES * 2);
      int mt  = rem >> 1;
      int nt  = rem & 1;
      int m   = mt * 16 + l16;
      const float* wchunk = wqkvf + (size_t)(((h * 3 + s) * 2 + nt) * 24) * 64
                                  + lane * 2;
      v8f acc = {};
      for (int kstep = 0; kstep < 24; ++kstep) {
        int k = kstep * 4 + 2 * half;
        v2f a, bf;
        int xa = xsidx(m, k);
        a.x = xs[xa];
        a.y = xs[xa + 1];
        bf  = *(const v2f*)(wchunk + kstep * 64);   // coalesced B-fragment
        acc = wmma4(a, bf, acc);
      }
      int col = nt * 16 + l16;
      float bb = bqkv[s * CDIM + h * HD + col];
      if (s == 0) {
        #pragma unroll
        for (int r = 0; r < 8; ++r)
          qs[(mt * 16 + r + 8 * half) * QSTR + col] = (acc[r] + bb) * scale;
      } else if (s == 1) {
        #pragma unroll
        for (int r = 0; r < 8; ++r)
          ks[(mt * 16 + r + 8 * half) * QSTR + col] = acc[r] + bb;
      } else {
        #pragma unroll
        for (int r = 0; r < 8; ++r)
          vs[(mt * 16 + r + 8 * half) * HD + col] = acc[r] + bb;
      }
    }
    __syncthreads();

    // ---- attention rows: 16-row tile per wave, online softmax, 22 col tiles
    for (int mt = wv; mt < NTILES; mt += 8) {
      v2f qf[8];
      #pragma unroll
      for (int k0 = 0; k0 < HD; k0 += 4) {
        int k = k0 + 2 * half;
        qf[k0 >> 2].x = qs[(mt * 16 + l16) * QSTR + k];
        qf[k0 >> 2].y = qs[(mt * 16 + l16) * QSTR + k + 1];
      }
      float rmax[8], rsum[8];
      #pragma unroll
      for (int r = 0; r < 8; ++r) { rmax[r] = -INFINITY; rsum[r] = 0.f; }
      v8f o0 = {}, o1 = {};

      for (int nt = 0; nt < NTILES; ++nt) {
        // S = q(mt) @ k(nt)^T   (K = 32, 8 wmma steps)
        v8f S = {};
        #pragma unroll
        for (int k0 = 0; k0 < HD; k0 += 4) {
          int k = k0 + 2 * half;
          v2f bf;
          bf.x = ks[(nt * 16 + l16) * QSTR + k];
          bf.y = ks[(nt * 16 + l16) * QSTR + k + 1];
          S = wmma4(qf[k0 >> 2], bf, S);
        }
        // precomputed bias + window mask + column bound (C layout)
        int n = nt * 16 + l16;
        #pragma unroll
        for (int r = 0; r < 8; ++r) {
          int m = mt * 16 + r + 8 * half;
          float sv = S[r];
          if (n >= NTOK) sv = -1.0e30f;
          else if (m >= 1 && m < NTOK && n >= 1) {
            int ij = (m - 1) * NX + (n - 1);
            sv += biasH[ij] + maskw[ij];
          }
          S[r] = sv;
        }
        // online softmax (reduce across the 16 lanes of each half)
        #pragma unroll
        for (int r = 0; r < 8; ++r) {
          float tmax = S[r];
          tmax = fmaxf(tmax, __shfl_xor(tmax, 1, 16));
          tmax = fmaxf(tmax, __shfl_xor(tmax, 2, 16));
          tmax = fmaxf(tmax, __shfl_xor(tmax, 4, 16));
          tmax = fmaxf(tmax, __shfl_xor(tmax, 8, 16));
          float nmax = fmaxf(rmax[r], tmax);
          float corr = __expf(rmax[r] - nmax);
          float p    = __expf(S[r]  - nmax);
          float ts = p;
          ts += __shfl_xor(ts, 1, 16);
          ts += __shfl_xor(ts, 2, 16);
          ts += __shfl_xor(ts, 4, 16);
          ts += __shfl_xor(ts, 8, 16);
          rsum[r] = rsum[r] * corr + ts;
          rmax[r] = nmax;
          S[r]    = p;
          o0[r]  *= corr;
          o1[r]  *= corr;
        }
        // transpose P (C layout -> A layout) via per-wave LDS scratch
        float* ps = pscr[wv];
        #pragma unroll
        for (int r = 0; r < 8; ++r)
          ps[(r + 8 * half) * 17 + l16] = S[r];
        asm volatile("s_wait_dscnt 0x0" ::: "memory");
        // O += P @ v(nt)   (K = 16, 4 wmma steps x 2 output col tiles)
        #pragma unroll
        for (int k0 = 0; k0 < 16; k0 += 4) {
          int kk = k0 + 2 * half;
          v2f a, b0, b1;
          a.x = ps[l16 * 17 + kk];
          a.y = ps[l16 * 17 + kk + 1];
          int vrow = nt * 16 + kk;
          b0.x = vs[vrow * HD + l16];
          b0.y = vs[(vrow + 1) * HD + l16];
          b1.x = vs[vrow * HD + 16 + l16];
          b1.y = vs[(vrow + 1) * HD + 16 + l16];
          o0 = wmma4(a, b0, o0);
          o1 = wmma4(a, b1, o1);
        }
        asm volatile("s_wait_dscnt 0x0" ::: "memory");
      }
      // normalize and write (B,344,96) workspace, head column slice
      #pragma unroll
      for (int r = 0; r < 8; ++r) {
        int m = mt * 16 + r + 8 * half;
        if (m < NTOK) {
          float inv = 1.f / rsum[r];
          size_t base = ((size_t)b * NTOK + m) * CDIM + h * HD;
          attn_out[base + l16]      = o0[r] * inv;
          attn_out[base + 16 + l16] = o1[r] * inv;
        }
      }
    }
    __syncthreads();   // qs/ks/vs reused by next head
  }
}

// ---------------------------------------------------------------------------
// Kernel 2: out = attn_ws(M x 96) @ Wproj^T + b_proj, scattered into
// x_out (rows n>=1) and gt_out (row n==0).  M = 512*344 = 1376 * 128 exactly.
// Input block staged via TDM; Wproj comes pre-fragmented (coalesced b64).
// ---------------------------------------------------------------------------
__global__ __launch_bounds__(256, 1)
void wattn_proj_kernel(const float* __restrict__ ain,
                       const float* __restrict__ wprojf,
                       const float* __restrict__ bp, float* __restrict__ out)
{
  __shared__ float as[128 * CDIM];     // 48KB input block (TDM, contiguous)
  const int tid  = threadIdx.x;
  const int lane = tid & 31;
  const int wv   = tid >> 5;
  const int half = lane >> 4;
  const int l16  = lane & 15;
  const size_t row0 = (size_t)blockIdx.x * 128;

  if (wv == 0) {
    tdm_load_1d((uint32_t)(uintptr_t)as, ain + row0 * CDIM,
                (uint32_t)(128 * CDIM), 0x00020000u);   // data_size=4B, no pad
    __builtin_amdgcn_s_wait_tensorcnt((short)0);
  }
  __syncthreads();

  v8f zero = {};
  v8f acc[6];
  #pragma unroll
  for (int nt = 0; nt < 6; ++nt) acc[nt] = zero;

  const float* wbase = wprojf + lane * 2;
  for (int kstep = 0; kstep < 24; ++kstep) {
    int k = kstep * 4 + 2 * half;
    v2f a;
    a.x = as[(wv * 16 + l16) * CDIM + k];
    a.y = as[(wv * 16 + l16) * CDIM + k + 1];
    #pragma unroll
    for (int nt = 0; nt < 6; ++nt) {
      v2f bf = *(const v2f*)(wbase + (nt * 24 + kstep) * 64);
      acc[nt] = wmma4(a, bf, acc[nt]);
    }
  }

  #pragma unroll
  for (int nt = 0; nt < 6; ++nt) {
    float bb = bp[nt * 16 + l16];
    int col  = nt * 16 + l16;
    #pragma unroll
    for (int r = 0; r < 8; ++r) {
      size_t gr = row0 + (size_t)wv * 16 + r + 8 * half;
      int bidx  = (int)(gr / NTOK);
      int n     = (int)(gr - (size_t)bidx * NTOK);
      float val = acc[nt][r] + bb;
      if (n == 0)
        out[X_OUT_ELEMS + (size_t)bidx * CDIM + col] = val;        // gt_out
      else
        out[((size_t)bidx * NX + (n - 1)) * CDIM + col] = val;     // x_out
    }
  }
}

extern "C" void kernel_launch(void* const* d_in, const int* in_sizes, int n_in,
                              void* d_out, int out_size, void* d_ws, size_t ws_size,
                              hipStream_t stream) {
  (void)in_sizes; (void)n_in; (void)out_size; (void)ws_size;
  const float* x     = (const float*)d_in[0];
  const float* gt    = (const float*)d_in[1];
  const float* mask  = (const float*)d_in[2];
  const int*   rel   = (const int*)d_in[3];
  const float* Wqkv  = (const float*)d_in[4];
  const float* bqkv  = (const float*)d_in[5];
  const float* Wproj = (const float*)d_in[6];
  const float* bproj = (const float*)d_in[7];
  const float* btab  = (const float*)d_in[8];

  float* wsf     = (float*)d_ws;
  float* attn_ws = wsf + ATTN_WS;
  float* biasHM  = wsf + BIAS_OFF;
  float* wqkvf   = wsf + WQKVF_OFF;
  float* wprojf  = wsf + WPROJF_OFF;

  wattn_prep_kernel<<<dim3((NXNX + 255) / 256), dim3(256), 0, stream>>>(
      rel, btab, Wqkv, Wproj, biasHM, wqkvf, wprojf);
  wattn_fused_kernel<<<dim3(512), dim3(256), 0, stream>>>(
      x, gt, mask, bqkv, biasHM, wqkvf, attn_ws);
  wattn_proj_kernel<<<dim3(1376), dim3(256), 0, stream>>>(
      attn_ws, wprojf, bproj, (float*)d_out);
}